// GIN_20005957664840
// MI455X (gfx1250) — compile-verified
//
#include <hip/hip_runtime.h>

#define IN_DIM 128
#define HID 64

typedef __attribute__((ext_vector_type(2))) float v2f;
typedef __attribute__((ext_vector_type(8))) float v8f;

// ---------------------------------------------------------------------------
// Zero a buffer (float4 granularity; all sizes here are multiples of 4 floats)
// ---------------------------------------------------------------------------
__global__ void gin_zero_kernel(float4* __restrict__ p, int n4) {
  int i = blockIdx.x * blockDim.x + threadIdx.x;
  if (i < n4) p[i] = make_float4(0.f, 0.f, 0.f, 0.f);
}

// ---------------------------------------------------------------------------
// Edge scatter-add: agg[dst[e]][:] += h[src[e]][:]
// One thread per (edge, float4 chunk). dim = 4 << logv floats per node.
// Non-returning atomicAdd lowers to global_atomic_add_f32 (L2 atomics).
// ---------------------------------------------------------------------------
__global__ void gin_scatter_kernel(const float* __restrict__ h,
                                   const int* __restrict__ src,
                                   const int* __restrict__ dst,
                                   float* __restrict__ agg,
                                   int logv, int total) {
  int i = blockIdx.x * blockDim.x + threadIdx.x;
  if (i >= total) return;
  int e = i >> logv;
  int v = i & ((1 << logv) - 1);
  int dim = 4 << logv;
  const float4 val = *(const float4*)(h + (size_t)src[e] * dim + v * 4);
  float* ap = agg + (size_t)dst[e] * dim + v * 4;
  atomicAdd(ap + 0, val.x);
  atomicAdd(ap + 1, val.y);
  atomicAdd(ap + 2, val.z);
  atomicAdd(ap + 3, val.w);
}

// ---------------------------------------------------------------------------
// Fused GEMM + bias + activation using V_WMMA_F32_16X16X4_F32.
//   Out(nRows x 64) = act( (A [+ AGG]) @ W(K x 64) + bias )
// One wave computes a 16-row x 64-col tile (4 accumulators of v8f).
// A-frag (16x4 f32): lane%16 = M row; lanes 0-15 hold K={0,1}, 16-31 K={2,3}.
// B-frag (4x16 f32): lane%16 = N col; lanes 0-15 hold K={0,1}, 16-31 K={2,3}.
// C/D (16x16 f32):   VGPR j holds M=j (lanes 0-15) / M=j+8 (lanes 16-31).
// ---------------------------------------------------------------------------
template <int K, bool ADD_AGG, bool LEAKY>
__global__ __launch_bounds__(256) void gin_gemm_kernel(
    const float* __restrict__ A, const float* __restrict__ AGG,
    const float* __restrict__ W,     // K x HID, row-major
    const float* __restrict__ bias,  // HID
    float* __restrict__ Out,         // nRows x HID
    int nRows) {
  const int lane = threadIdx.x & 31;
  const int wave = threadIdx.x >> 5;
  const int rowBase = (blockIdx.x * (blockDim.x >> 5) + wave) * 16;
  if (rowBase >= nRows) return;  // wave-uniform: EXEC stays all-ones for WMMA

  const int half = lane >> 4;  // 0: K={0,1}, 1: K={2,3} within each k-step
  const int l16 = lane & 15;
  const int row = rowBase + l16;

  const float* arow = A + (size_t)row * K + 2 * half;
  const float* grow = ADD_AGG ? (AGG + (size_t)row * K + 2 * half) : nullptr;

  v8f acc0 = {}, acc1 = {}, acc2 = {}, acc3 = {};

#pragma unroll 4
  for (int kk = 0; kk < K; kk += 4) {
    v2f a = *(const v2f*)(arow + kk);
    if constexpr (ADD_AGG) {
      v2f g = *(const v2f*)(grow + kk);
      a += g;
    }
    const float* wp = W + (size_t)(kk + 2 * half) * HID + l16;
    v2f b0, b1, b2, b3;
    b0.x = wp[0];
    b0.y = wp[HID];
    b1.x = wp[16];
    b1.y = wp[16 + HID];
    b2.x = wp[32];
    b2.y = wp[32 + HID];
    b3.x = wp[48];
    b3.y = wp[48 + HID];
    acc0 = __builtin_amdgcn_wmma_f32_16x16x4_f32(false, a, false, b0, (short)0,
                                                 acc0, false, false);
    acc1 = __builtin_amdgcn_wmma_f32_16x16x4_f32(false, a, false, b1, (short)0,
                                                 acc1, false, false);
    acc2 = __builtin_amdgcn_wmma_f32_16x16x4_f32(false, a, false, b2, (short)0,
                                                 acc2, false, false);
    acc3 = __builtin_amdgcn_wmma_f32_16x16x4_f32(false, a, false, b3, (short)0,
                                                 acc3, false, false);
  }

  const float bias0 = bias[l16];
  const float bias1 = bias[16 + l16];
  const float bias2 = bias[32 + l16];
  const float bias3 = bias[48 + l16];

  auto act = [](float x) {
    if constexpr (LEAKY)
      return x > 0.f ? x : 0.01f * x;
    else
      return fmaxf(x, 0.f);
  };

#pragma unroll
  for (int j = 0; j < 8; ++j) {
    const int r = rowBase + j + 8 * half;
    float* op = Out + (size_t)r * HID + l16;
    op[0] = act(acc0[j] + bias0);
    op[16] = act(acc1[j] + bias1);
    op[32] = act(acc2[j] + bias2);
    op[48] = act(acc3[j] + bias3);
  }
}

// ---------------------------------------------------------------------------
// Host-side launch. Workspace layout (floats):
//   agg : [0, N*128)           (layer 2 reuses first N*64)
//   z   : [N*128, N*128+N*64)  (MLP hidden, both layers)
// d_out doubles as layer-1 output h1 (overwritten last by layer-2 GEMM).
// ---------------------------------------------------------------------------
extern "C" void kernel_launch(void* const* d_in, const int* in_sizes, int n_in,
                              void* d_out, int out_size, void* d_ws,
                              size_t ws_size, hipStream_t stream) {
  const float* h = (const float*)d_in[0];
  const int* ei = (const int*)d_in[1];
  const float* W1a = (const float*)d_in[2];
  const float* b1a = (const float*)d_in[3];
  const float* W1b = (const float*)d_in[4];
  const float* b1b = (const float*)d_in[5];
  const float* W2a = (const float*)d_in[6];
  const float* b2a = (const float*)d_in[7];
  const float* W2b = (const float*)d_in[8];
  const float* b2b = (const float*)d_in[9];

  const int N = in_sizes[0] / IN_DIM;
  const int E = in_sizes[1] / 2;
  const int* src = ei;
  const int* dst = ei + E;

  float* agg = (float*)d_ws;
  float* z = agg + (size_t)N * IN_DIM;
  float* h1 = (float*)d_out;
  float* out = (float*)d_out;

  const int rowTiles = (N + 15) / 16;
  const int gemmBlocks = (rowTiles + 7) / 8;  // 8 waves (16-row tiles) / block

  // ---- Layer 1 ----
  {
    int n4 = (N * IN_DIM) / 4;
    gin_zero_kernel<<<(n4 + 255) / 256, 256, 0, stream>>>((float4*)agg, n4);
    int total = E * (IN_DIM / 4);  // logv = 5 (32 float4 chunks / node)
    gin_scatter_kernel<<<(total + 255) / 256, 256, 0, stream>>>(h, src, dst,
                                                                agg, 5, total);
    gin_gemm_kernel<IN_DIM, true, false>
        <<<gemmBlocks, 256, 0, stream>>>(h, agg, W1a, b1a, z, N);
    gin_gemm_kernel<HID, false, true>
        <<<gemmBlocks, 256, 0, stream>>>(z, nullptr, W1b, b1b, h1, N);
  }

  // ---- Layer 2 ----
  {
    int n4 = (N * HID) / 4;
    gin_zero_kernel<<<(n4 + 255) / 256, 256, 0, stream>>>((float4*)agg, n4);
    int total = E * (HID / 4);  // logv = 4 (16 float4 chunks / node)
    gin_scatter_kernel<<<(total + 255) / 256, 256, 0, stream>>>(h1, src, dst,
                                                                agg, 4, total);
    gin_gemm_kernel<HID, true, false>
        <<<gemmBlocks, 256, 0, stream>>>(h1, agg, W2a, b2a, z, N);
    gin_gemm_kernel<HID, false, true>
        <<<gemmBlocks, 256, 0, stream>>>(z, nullptr, W2b, b2b, out, N);
  }
}